// ImportancePropagationLayer_71459665871598
// MI455X (gfx1250) — compile-verified
//
#include <hip/hip_runtime.h>
#include <hip/hip_bf16.h>

typedef __attribute__((ext_vector_type(2))) float v2f;
typedef __attribute__((ext_vector_type(8))) float v8f;

#define OC 128
#define HEADS 8
#define CH 16
#define EDIM 16
#define NEG_SLOPE 0.2f

static __device__ __forceinline__ v8f wmma_f32_k4(v2f a, v2f b, v8f c) {
    // D = A(16x4 f32) x B(4x16 f32) + C(16x16 f32), exact fp32
    return __builtin_amdgcn_wmma_f32_16x16x4_f32(false, a, false, b, (short)0, c,
                                                 false, false);
}

static __device__ __forceinline__ void atomicMaxF(float* addr, float v) {
    if (v >= 0.0f) {
        atomicMax((int*)addr, __float_as_int(v));
    } else {
        atomicMin((unsigned int*)addr, __float_as_uint(v));
    }
}

// ---------------------------------------------------------------------------
// Kernel 1: x_l = x@Wl + bl ; x_r = x@Wr + br   (blockIdx.y selects l/r)
// One wave computes a 16-row x 128-col tile via WMMA f32 16x16x4.
// B fragments for all 8 column tiles are loaded first (one clause), then the
// 8 independent WMMAs issue back-to-back.
// ---------------------------------------------------------------------------
__global__ void node_gemm_kernel(const float* __restrict__ x,
                                 const float* __restrict__ Wl, const float* __restrict__ bl,
                                 const float* __restrict__ Wr, const float* __restrict__ br,
                                 float* __restrict__ xl, float* __restrict__ xr, int N) {
    const float* W    = blockIdx.y ? Wr : Wl;
    const float* bias = blockIdx.y ? br : bl;
    float*       outp = blockIdx.y ? xr : xl;

    const int  lane = threadIdx.x;
    const int  lr   = lane & 15;
    const int  hi   = lane >> 4;
    const long rowbase = (long)blockIdx.x * 16;
    long arow = rowbase + lr;
    if (arow >= N) arow = N - 1;

    const float* xrow = x + arow * OC;

    v8f acc[8];
#pragma unroll
    for (int i = 0; i < 8; ++i) acc[i] = (v8f){0, 0, 0, 0, 0, 0, 0, 0};

    for (int kk = 0; kk < OC; kk += 4) {
        const int kb = kk + 2 * hi;
        v2f a;
        a.x = xrow[kb];
        a.y = xrow[kb + 1];
        v2f b[8];
#pragma unroll
        for (int nt = 0; nt < 8; ++nt) {
            const int col = nt * 16 + lr;
            b[nt].x = W[(long)kb * OC + col];
            b[nt].y = W[(long)(kb + 1) * OC + col];
        }
#pragma unroll
        for (int nt = 0; nt < 8; ++nt) acc[nt] = wmma_f32_k4(a, b[nt], acc[nt]);
    }
#pragma unroll
    for (int nt = 0; nt < 8; ++nt) {
        const int   col = nt * 16 + lr;
        const float bv  = bias[col];
#pragma unroll
        for (int v = 0; v < 8; ++v) {
            const long orow = rowbase + v + 8 * hi;
            if (orow < N) outp[orow * OC + col] = acc[nt][v] + bv;
        }
    }
}

// ---------------------------------------------------------------------------
// Kernel 2: initialize scratch (agg=0, m=-inf, denom=0)
// ---------------------------------------------------------------------------
__global__ void fill_kernel(float* __restrict__ agg, float* __restrict__ m,
                            float* __restrict__ denom, long n_agg, long n_mh) {
    const long i      = (long)blockIdx.x * blockDim.x + threadIdx.x;
    const long stride = (long)gridDim.x * blockDim.x;
    for (long j = i; j < n_agg; j += stride) agg[j] = 0.0f;
    for (long j = i; j < n_mh; j += stride) {
        m[j]     = -3.0e38f;
        denom[j] = 0.0f;
    }
}

// ---------------------------------------------------------------------------
// Kernel 3: e = edge_attr@We via WMMA -> LDS, then fused
//           logits[e,h] = att[h] . leakyrelu(e + x_l[src] + x_r[dst]),
//           atomic max into m[dst,h].
// Block = 128 threads (4 waves). Each wave computes 2 of the 8 column tiles
// of the 16x128 e-tile (same A fragment, different B), so all waves do WMMA.
// ---------------------------------------------------------------------------
__global__ void edge_logits_kernel(const float* __restrict__ ea, const float* __restrict__ We,
                                   const int* __restrict__ ei,
                                   const float* __restrict__ xl, const float* __restrict__ xr,
                                   const float* __restrict__ att,
                                   float* __restrict__ logits, float* __restrict__ m,
                                   int E) {
    __shared__ float etile[16 * OC];
    const long ebase = (long)blockIdx.x * 16;
    const int  tid   = threadIdx.x;

    {
        const int wv   = tid >> 5;   // wave 0..3 -> column tiles 2wv, 2wv+1
        const int lane = tid & 31;
        const int lr   = lane & 15;
        const int hi   = lane >> 4;
        long ar = ebase + lr;
        if (ar >= E) ar = E - 1;
        const float* earow = ea + ar * EDIM;

        v8f acc[2];
#pragma unroll
        for (int i = 0; i < 2; ++i) acc[i] = (v8f){0, 0, 0, 0, 0, 0, 0, 0};
#pragma unroll
        for (int kk = 0; kk < EDIM; kk += 4) {
            const int kb = kk + 2 * hi;
            v2f a;
            a.x = earow[kb];
            a.y = earow[kb + 1];
            v2f b[2];
#pragma unroll
            for (int t = 0; t < 2; ++t) {
                const int col = (wv * 2 + t) * 16 + lr;
                b[t].x = We[kb * OC + col];
                b[t].y = We[(kb + 1) * OC + col];
            }
#pragma unroll
            for (int t = 0; t < 2; ++t) acc[t] = wmma_f32_k4(a, b[t], acc[t]);
        }
#pragma unroll
        for (int t = 0; t < 2; ++t)
#pragma unroll
            for (int v = 0; v < 8; ++v)
                etile[(v + 8 * hi) * OC + (wv * 2 + t) * 16 + lr] = acc[t][v];
    }
    __syncthreads();

    const int  el = tid >> 3;   // 0..15 local edge
    const int  h  = tid & 7;    // head
    const long e  = ebase + el;
    if (e < E) {
        const long s = ei[e];
        const long d = ei[(long)E + e];
        const float4* ep = (const float4*)&etile[el * OC + h * CH];
        const float4* xs = (const float4*)&xl[s * OC + h * CH];
        const float4* xd = (const float4*)&xr[d * OC + h * CH];
        const float4* av = (const float4*)&att[h * CH];
        float lg = 0.0f;
#pragma unroll
        for (int q = 0; q < 4; ++q) {
            const float4 ev = ep[q], sv = xs[q], dv = xd[q], aw = av[q];
            float v0 = ev.x + sv.x + dv.x;
            float v1 = ev.y + sv.y + dv.y;
            float v2 = ev.z + sv.z + dv.z;
            float v3 = ev.w + sv.w + dv.w;
            v0 = v0 > 0.0f ? v0 : NEG_SLOPE * v0;
            v1 = v1 > 0.0f ? v1 : NEG_SLOPE * v1;
            v2 = v2 > 0.0f ? v2 : NEG_SLOPE * v2;
            v3 = v3 > 0.0f ? v3 : NEG_SLOPE * v3;
            lg += v0 * aw.x + v1 * aw.y + v2 * aw.z + v3 * aw.w;
        }
        logits[e * HEADS + h] = lg;
        atomicMaxF(&m[d * HEADS + h], lg);
    }
}

// ---------------------------------------------------------------------------
// Kernel 4: p = exp(logit - m[dst]); denom[dst] += p  (p overwrites logits)
// ---------------------------------------------------------------------------
__global__ void edge_exp_kernel(const int* __restrict__ ei, float* __restrict__ logits,
                                const float* __restrict__ m, float* __restrict__ denom,
                                int E) {
    const long i = (long)blockIdx.x * blockDim.x + threadIdx.x;
    if (i >= (long)E * HEADS) return;
    const long e = i / HEADS;
    const int  h = (int)(i % HEADS);
    const long d = ei[(long)E + e];
    const float p = expf(logits[i] - m[d * HEADS + h]);
    logits[i] = p;
    atomicAdd(&denom[d * HEADS + h], p);
}

// ---------------------------------------------------------------------------
// Kernel 5: agg[dst] += x_l[src] * (p / denom[dst])  (per head, 16 ch each)
// ---------------------------------------------------------------------------
__global__ void edge_scatter_kernel(const int* __restrict__ ei, const float* __restrict__ p,
                                    const float* __restrict__ denom,
                                    const float* __restrict__ xl, float* __restrict__ agg,
                                    int E) {
    const long i = (long)blockIdx.x * blockDim.x + threadIdx.x;
    if (i >= (long)E * HEADS) return;
    const long e = i / HEADS;
    const int  h = (int)(i % HEADS);
    const long s = ei[e];
    const long d = ei[(long)E + e];
    const float alpha = p[i] / denom[d * HEADS + h];
    const float4* xs = (const float4*)&xl[s * OC + h * CH];
    float*        ag = &agg[d * OC + h * CH];
#pragma unroll
    for (int q = 0; q < 4; ++q) {
        const float4 v = xs[q];
        atomicAdd(&ag[q * 4 + 0], v.x * alpha);
        atomicAdd(&ag[q * 4 + 1], v.y * alpha);
        atomicAdd(&ag[q * 4 + 2], v.z * alpha);
        atomicAdd(&ag[q * 4 + 3], v.w * alpha);
    }
}

// ---------------------------------------------------------------------------
// Kernel 6: conv = agg + conv_bias; gate = sigmoid([conv,imp]@Wg + bg);
//           out = gate*conv + (1-gate)*x; prop = out@Wp + bp.
//           Gate GEMM via WMMA (K=128 of conv) + rank-1 importance term.
// ---------------------------------------------------------------------------
__global__ void gate_kernel(const float* __restrict__ agg, const float* __restrict__ conv_bias,
                            const float* __restrict__ importance, const float* __restrict__ x,
                            const float* __restrict__ Wg, const float* __restrict__ bg,
                            const float* __restrict__ Wp, const float* __restrict__ bp,
                            float* __restrict__ outp, float* __restrict__ prop, int N) {
    const int  lane = threadIdx.x;
    const int  lr   = lane & 15;
    const int  hi   = lane >> 4;
    const long rowbase = (long)blockIdx.x * 16;
    long arow = rowbase + lr;
    if (arow >= N) arow = N - 1;
    const float* aggrow = agg + arow * OC;

    v8f acc[8];
#pragma unroll
    for (int i = 0; i < 8; ++i) acc[i] = (v8f){0, 0, 0, 0, 0, 0, 0, 0};

    for (int kk = 0; kk < OC; kk += 4) {
        const int kb = kk + 2 * hi;
        v2f a;
        a.x = aggrow[kb] + conv_bias[kb];
        a.y = aggrow[kb + 1] + conv_bias[kb + 1];
        v2f b[8];
#pragma unroll
        for (int nt = 0; nt < 8; ++nt) {
            const int col = nt * 16 + lr;
            b[nt].x = Wg[(long)kb * OC + col];
            b[nt].y = Wg[(long)(kb + 1) * OC + col];
        }
#pragma unroll
        for (int nt = 0; nt < 8; ++nt) acc[nt] = wmma_f32_k4(a, b[nt], acc[nt]);
    }

    float imp[8];
#pragma unroll
    for (int v = 0; v < 8; ++v) {
        long r = rowbase + v + 8 * hi;
        if (r >= N) r = N - 1;
        imp[v] = importance[r];
    }

    float prop_part[8];
#pragma unroll
    for (int v = 0; v < 8; ++v) prop_part[v] = 0.0f;

#pragma unroll
    for (int nt = 0; nt < 8; ++nt) {
        const int   col = nt * 16 + lr;
        const float bgv = bg[col];
        const float cb  = conv_bias[col];
        const float wpv = Wp[col];
        const float wgl = Wg[(long)OC * OC + col];  // importance row of Wg (row 128)
#pragma unroll
        for (int v = 0; v < 8; ++v) {
            const long orow = rowbase + v + 8 * hi;
            const long cr   = orow < N ? orow : (long)N - 1;
            const float z    = acc[nt][v] + imp[v] * wgl + bgv;
            const float g    = 1.0f / (1.0f + expf(-z));
            const float conv = agg[cr * OC + col] + cb;
            const float xv   = x[cr * OC + col];
            const float o    = g * conv + (1.0f - g) * xv;
            if (orow < N) outp[orow * OC + col] = o;
            prop_part[v] += o * wpv;
        }
    }

    // reduce prop over the 16 lanes of each half-wave (cols 0..127 total)
#pragma unroll
    for (int mask = 1; mask < 16; mask <<= 1) {
#pragma unroll
        for (int v = 0; v < 8; ++v)
            prop_part[v] += __shfl_xor(prop_part[v], mask, 32);
    }
    if (lr == 0) {
        const float bpv = bp[0];
#pragma unroll
        for (int v = 0; v < 8; ++v) {
            const long orow = rowbase + v + 8 * hi;
            if (orow < N) prop[orow] = prop_part[v] + bpv;
        }
    }
}

// ---------------------------------------------------------------------------
extern "C" void kernel_launch(void* const* d_in, const int* in_sizes, int n_in,
                              void* d_out, int out_size, void* d_ws, size_t ws_size,
                              hipStream_t stream) {
    const float* x          = (const float*)d_in[0];
    const int*   ei         = (const int*)d_in[1];
    const float* ea         = (const float*)d_in[2];
    const float* importance = (const float*)d_in[3];
    const float* Wl         = (const float*)d_in[4];
    const float* bl         = (const float*)d_in[5];
    const float* Wr         = (const float*)d_in[6];
    const float* br         = (const float*)d_in[7];
    const float* We         = (const float*)d_in[8];
    const float* att        = (const float*)d_in[9];
    const float* conv_bias  = (const float*)d_in[10];
    const float* Wg         = (const float*)d_in[11];
    const float* bg         = (const float*)d_in[12];
    const float* Wp         = (const float*)d_in[13];
    const float* bp         = (const float*)d_in[14];

    const int N = in_sizes[0] / OC;
    const int E = in_sizes[2] / EDIM;

    float* ws     = (float*)d_ws;
    float* xl     = ws;
    float* xr     = xl + (size_t)N * OC;
    float* agg    = xr + (size_t)N * OC;
    float* m      = agg + (size_t)N * OC;
    float* denom  = m + (size_t)N * HEADS;
    float* logits = denom + (size_t)N * HEADS;  // later holds p

    float* outp = (float*)d_out;
    float* prop = outp + (size_t)N * OC;

    const int ntiles = (N + 15) / 16;
    const int etiles = (E + 15) / 16;

    node_gemm_kernel<<<dim3(ntiles, 2), 32, 0, stream>>>(x, Wl, bl, Wr, br, xl, xr, N);

    const long n_agg = (long)N * OC, n_mh = (long)N * HEADS;
    fill_kernel<<<2048, 256, 0, stream>>>(agg, m, denom, n_agg, n_mh);

    edge_logits_kernel<<<etiles, 128, 0, stream>>>(ea, We, ei, xl, xr, att, logits, m, E);

    const long eh = (long)E * HEADS;
    edge_exp_kernel<<<(int)((eh + 255) / 256), 256, 0, stream>>>(ei, logits, m, denom, E);
    edge_scatter_kernel<<<(int)((eh + 255) / 256), 256, 0, stream>>>(ei, logits, denom, xl, agg, E);

    gate_kernel<<<ntiles, 32, 0, stream>>>(agg, conv_bias, importance, x, Wg, bg, Wp, bp,
                                           outp, prop, N);
}